// Q_group_feb_39470749450588
// MI455X (gfx1250) — compile-verified
//
#include <hip/hip_runtime.h>
#include <hip/hip_bf16.h>

typedef __attribute__((ext_vector_type(2))) float v2f;
typedef __attribute__((ext_vector_type(8))) float v8f;
typedef int v4i_ __attribute__((vector_size(16)));

#define L_DIM 1024
#define B_DIM 16
#define X_DIM 512
#define Y_DIM 512
#define Q_DIM 64
#define NUM_D 4
#define H_LOW 128
#define MROWS (L_DIM * B_DIM)   // 16384

// ---- dynamic-LDS layout for the flash kernel (floats), rows padded for banks
#define SM_QI 0                       // [4][16][68]
#define SM_QJ 4352                    // [4][16][68]
#define SM_VT 8704                    // [16][520]
#define SM_PT (8704 + 16 * 520)       // [4][16][20]  -> 17024
#define SM_FLOATS (SM_PT + 4 * 16 * 20)   // 18304 floats = 73216 B

static __device__ __forceinline__ v8f zero8() {
    v8f z;
#pragma unroll
    for (int i = 0; i < 8; ++i) z[i] = 0.0f;
    return z;
}

// ---- CDNA5 async global->LDS staging (guarded; falls back to plain copies)
#if __has_builtin(__builtin_amdgcn_global_load_async_to_lds_b128)
#define HAVE_ASYNC_LDS 1
static __device__ __forceinline__ void cp16_async(const float* g, float* l) {
    __builtin_amdgcn_global_load_async_to_lds_b128(
        (__attribute__((address_space(1))) v4i_*)g,
        (__attribute__((address_space(3))) v4i_*)l, 0, 0);
}
static __device__ __forceinline__ void wait_async0() {
#if __has_builtin(__builtin_amdgcn_s_wait_asynccnt)
    __builtin_amdgcn_s_wait_asynccnt(0);
#else
    asm volatile("s_wait_asynccnt 0x0" ::: "memory");
#endif
}
#else
#define HAVE_ASYNC_LDS 0
#endif

static __device__ __forceinline__ void cp16(const float* g, float* l) {
#if HAVE_ASYNC_LDS
    cp16_async(g, l);
#else
    *(float4*)l = *(const float4*)g;
#endif
}
static __device__ __forceinline__ void cp_fence() {
#if HAVE_ASYNC_LDS
    wait_async0();
#endif
}

// ---------------------------------------------------------------------------
// C[M,N] = act(A[M,K] @ B[N,K]^T + bias[N]), f32 WMMA 16x16x4, software
// pipelined (double-buffered groups of 4 K-chunks). One wave per block,
// 16x64 C tile. Optional fused row-squared-norm output (Qins only).
// ---------------------------------------------------------------------------
__global__ __launch_bounds__(32) void wmma_gemm_bias_kernel(
    const float* __restrict__ A, const float* __restrict__ Bw,
    const float* __restrict__ bias, float* __restrict__ C,
    float* __restrict__ sqout, int M, int N, int K, int act_sigmoid)
{
    const int lane = threadIdx.x & 31;
    const int half = lane >> 4;
    const int l16  = lane & 15;
    const int m0 = blockIdx.x * 16;
    const int n0 = blockIdx.y * 64;

    v8f acc[4];
#pragma unroll
    for (int t = 0; t < 4; ++t) acc[t] = zero8();

    const float* arow = A + (size_t)(m0 + l16) * K + 2 * half;
    const float* brow[4];
#pragma unroll
    for (int t = 0; t < 4; ++t)
        brow[t] = Bw + (size_t)(n0 + t * 16 + l16) * K + 2 * half;

    v2f a0[4], a1[4], b0[4][4], b1[4][4];

    auto LOADG = [&](int g, v2f (&aa)[4], v2f (&bb)[4][4]) {
        const int kb = g * 16;
#pragma unroll
        for (int c = 0; c < 4; ++c) {
            aa[c] = *(const v2f*)(arow + kb + 4 * c);
#pragma unroll
            for (int t = 0; t < 4; ++t)
                bb[c][t] = *(const v2f*)(brow[t] + kb + 4 * c);
        }
    };
    auto COMPG = [&](v2f (&aa)[4], v2f (&bb)[4][4]) {
#pragma unroll
        for (int c = 0; c < 4; ++c)
#pragma unroll
            for (int t = 0; t < 4; ++t)
                acc[t] = __builtin_amdgcn_wmma_f32_16x16x4_f32(
                    false, aa[c], false, bb[c][t], (short)0, acc[t], false, false);
    };

    const int ng = K >> 4;           // 32 (K=512) or 8 (K=128); always even
    LOADG(0, a0, b0);
    for (int g = 0; g < ng; g += 2) {
        LOADG(g + 1, a1, b1);
        COMPG(a0, b0);
        if (g + 2 < ng) LOADG(g + 2, a0, b0);
        COMPG(a1, b1);
    }

    float bv[4];
#pragma unroll
    for (int t = 0; t < 4; ++t) bv[t] = bias[n0 + t * 16 + l16];

#pragma unroll
    for (int r = 0; r < 8; ++r) {
        float ss = 0.0f;
#pragma unroll
        for (int t = 0; t < 4; ++t) {
            float v = acc[t][r] + bv[t];
            if (act_sigmoid) v = 1.0f / (1.0f + __expf(-v));
            C[(size_t)(m0 + r + 8 * half) * N + (n0 + t * 16 + l16)] = v;
            ss += v * v;
        }
        if (sqout) {   // Qins call: n0 = d*64, m0 = l*16, row b = r+8*half
            ss += __shfl_xor(ss, 1);
            ss += __shfl_xor(ss, 2);
            ss += __shfl_xor(ss, 4);
            ss += __shfl_xor(ss, 8);
            if (l16 == 0) {
                int bb_ = r + 8 * half;
                sqout[(((n0 >> 6) * 16 + bb_) << 10) + (m0 >> 4)] = ss;
            }
        }
    }
}

// ---------------------------------------------------------------------------
// temp_Qins[b][q] = mean over (d,l) of Qins
// ---------------------------------------------------------------------------
__global__ __launch_bounds__(64) void tempq_kernel(const float* __restrict__ Qrm,
                                                   float* __restrict__ out2)
{
    int b = blockIdx.x;
    int q = threadIdx.x;
    float s = 0.0f;
    for (int d = 0; d < NUM_D; ++d)
        for (int l = 0; l < L_DIM; ++l)
            s += Qrm[((size_t)l * 16 + b) * (NUM_D * Q_DIM) + d * Q_DIM + q];
    out2[b * Q_DIM + q] = s * (1.0f / (NUM_D * L_DIM));
}

// ---------------------------------------------------------------------------
// Fused: signed-L2 scores -> +mask -> online softmax -> @ Sin (flash style).
// grid = (L/16, B); block = 128 (4 waves). Wave w owns Y columns [w*128,+128).
// Q_i, Q_j, V tiles staged in (dynamic) LDS; P re-fragmented through LDS.
// ---------------------------------------------------------------------------
__global__ __launch_bounds__(128) void flash_kernel(
    const float* __restrict__ Qrm, const float* __restrict__ sqb,
    const float* __restrict__ Sin, const float* __restrict__ mask,
    float* __restrict__ Sout)
{
    extern __shared__ float sm[];

    const int tid  = threadIdx.x;
    const int lane = tid & 31;
    const int w    = tid >> 5;
    const int half = lane >> 4;
    const int l16  = lane & 15;
    const int i0   = blockIdx.x * 16;
    const int b    = blockIdx.y;
    const int y0   = w * 128;

    // --- stage Q_i tile once: 1024 float4 chunks (row pad 68) ---
#pragma unroll
    for (int it = 0; it < 8; ++it) {
        int f = tid + it * 128;
        int d = f >> 8, rem = f & 255, r = rem >> 4, v4 = rem & 15;
        cp16(&Qrm[((size_t)(i0 + r) * 16 + b) * 256 + d * 64 + 4 * v4],
             &sm[SM_QI + d * 1088 + r * 68 + 4 * v4]);
    }
    cp_fence();
    __syncthreads();

    float sqiv[NUM_D][8];
#pragma unroll
    for (int d = 0; d < NUM_D; ++d)
#pragma unroll
        for (int r = 0; r < 8; ++r)
            sqiv[d][r] = sqb[((d * 16 + b) << 10) + i0 + r + 8 * half];

    float mrun[8], lrun[8];
#pragma unroll
    for (int r = 0; r < 8; ++r) { mrun[r] = -3.0e38f; lrun[r] = 0.0f; }
    v8f O[8];
#pragma unroll
    for (int t = 0; t < 8; ++t) O[t] = zero8();

    const float coeff[NUM_D] = {1.0f, -1.0f, 1.0f, -1.0f};
    const float invs = 0.04419417382415922f;   // 1/sqrt(512)

    for (int j0 = 0; j0 < L_DIM; j0 += 16) {
        __syncthreads();
        // stage Q_j (1024 f4) and V = Sin j-tile (2048 f4, row pad 520)
#pragma unroll
        for (int it = 0; it < 8; ++it) {
            int f = tid + it * 128;
            int d = f >> 8, rem = f & 255, r = rem >> 4, v4 = rem & 15;
            cp16(&Qrm[((size_t)(j0 + r) * 16 + b) * 256 + d * 64 + 4 * v4],
                 &sm[SM_QJ + d * 1088 + r * 68 + 4 * v4]);
        }
#pragma unroll
        for (int it = 0; it < 16; ++it) {
            int f = tid + it * 128;
            int r = f >> 7, v4 = f & 127;
            cp16(&Sin[((size_t)(j0 + r) * 16 + b) * Y_DIM + 4 * v4],
                 &sm[SM_VT + r * 520 + 4 * v4]);
        }
        cp_fence();
        __syncthreads();

        // ---- scores: s = sum_d coeff_d * sqrt(max(sqi + sqj - 2*G_d, 0)) ----
        float s[8];
#pragma unroll
        for (int r = 0; r < 8; ++r) s[r] = 0.0f;
#pragma unroll
        for (int d = 0; d < NUM_D; ++d) {
            v8f g = zero8();
            const float* qiB = &sm[SM_QI + d * 1088 + l16 * 68 + 2 * half];
            const float* qjB = &sm[SM_QJ + d * 1088 + l16 * 68 + 2 * half];
#pragma unroll
            for (int grp = 0; grp < 4; ++grp) {
                v2f af[4], bf[4];
#pragma unroll
                for (int c = 0; c < 4; ++c) {
                    af[c] = *(const v2f*)(qiB + grp * 16 + 4 * c);
                    bf[c] = *(const v2f*)(qjB + grp * 16 + 4 * c);
                }
#pragma unroll
                for (int c = 0; c < 4; ++c)
                    g = __builtin_amdgcn_wmma_f32_16x16x4_f32(
                        false, af[c], false, bf[c], (short)0, g, false, false);
            }
            float sqj = sqb[((d * 16 + b) << 10) + j0 + l16];
#pragma unroll
            for (int r = 0; r < 8; ++r) {
                float d2 = fmaxf(sqiv[d][r] + sqj - 2.0f * g[r], 0.0f);
                s[r] += coeff[d] * sqrtf(d2);
            }
        }

        // ---- +mask, online softmax (row ops = 16-lane butterflies) ----
#pragma unroll
        for (int r = 0; r < 8; ++r) {
            float sv = s[r] * invs +
                       mask[(size_t)(i0 + r + 8 * half) * L_DIM + j0 + l16];
            float mx = sv;
            mx = fmaxf(mx, __shfl_xor(mx, 1));
            mx = fmaxf(mx, __shfl_xor(mx, 2));
            mx = fmaxf(mx, __shfl_xor(mx, 4));
            mx = fmaxf(mx, __shfl_xor(mx, 8));
            float mnew  = fmaxf(mrun[r], mx);
            float alpha = __expf(mrun[r] - mnew);
            float p     = __expf(sv - mnew);
            float rs = p;
            rs += __shfl_xor(rs, 1);
            rs += __shfl_xor(rs, 2);
            rs += __shfl_xor(rs, 4);
            rs += __shfl_xor(rs, 8);
            lrun[r] = lrun[r] * alpha + rs;
            mrun[r] = mnew;
#pragma unroll
            for (int t = 0; t < 8; ++t) O[t][r] *= alpha;   // same layout
            sm[SM_PT + w * 320 + (r + 8 * half) * 20 + l16] = p;
        }

        // ---- O += P @ V  (K = 16 -> 4 WMMA chunks, B frags from LDS) ----
#pragma unroll
        for (int c = 0; c < 4; ++c) {
            v2f a = *(const v2f*)&sm[SM_PT + w * 320 + l16 * 20 + 4 * c + 2 * half];
            const int k = 4 * c + 2 * half;
            const float* vrow = &sm[SM_VT + k * 520 + y0];
            v2f bf[8];
#pragma unroll
            for (int t = 0; t < 8; ++t) {
                bf[t].x = vrow[t * 16 + l16];
                bf[t].y = vrow[520 + t * 16 + l16];
            }
#pragma unroll
            for (int t = 0; t < 8; ++t)
                O[t] = __builtin_amdgcn_wmma_f32_16x16x4_f32(
                    false, a, false, bf[t], (short)0, O[t], false, false);
        }
    }

    // ---- normalize and store Sout[b][i][y] ----
#pragma unroll
    for (int t = 0; t < 8; ++t) {
#pragma unroll
        for (int r = 0; r < 8; ++r) {
            Sout[((size_t)b * L_DIM + i0 + r + 8 * half) * Y_DIM + y0 + t * 16 + l16] =
                O[t][r] / lrun[r];
        }
    }
}

// ---------------------------------------------------------------------------
extern "C" void kernel_launch(void* const* d_in, const int* in_sizes, int n_in,
                              void* d_out, int out_size, void* d_ws, size_t ws_size,
                              hipStream_t stream) {
    const float* X    = (const float*)d_in[0];
    const float* mask = (const float*)d_in[1];
    const float* Wq   = (const float*)d_in[2];
    const float* bq   = (const float*)d_in[3];
    const float* Wh   = (const float*)d_in[4];
    const float* bh   = (const float*)d_in[5];
    const float* Ws   = (const float*)d_in[6];
    const float* bs   = (const float*)d_in[7];
    float* out = (float*)d_out;

    // workspace layout (f32): Qrm 16MB | h 8MB | Sin 32MB | sq 256KB
    char* ws = (char*)d_ws;
    float* Qrm  = (float*)(ws);
    float* hbuf = (float*)(ws + (size_t)(16 << 20));
    float* Sinb = (float*)(ws + (size_t)(24 << 20));
    float* sqb  = (float*)(ws + (size_t)(56 << 20));

    // Qins (row-major [(l*16+b) x 256], col = d*64+q) + fused sq
    wmma_gemm_bias_kernel<<<dim3(MROWS / 16, (NUM_D * Q_DIM) / 64), 32, 0, stream>>>(
        X, Wq, bq, Qrm, sqb, MROWS, NUM_D * Q_DIM, X_DIM, 0);
    // h = sigmoid(X Wh^T + bh)
    wmma_gemm_bias_kernel<<<dim3(MROWS / 16, H_LOW / 64), 32, 0, stream>>>(
        X, Wh, bh, hbuf, nullptr, MROWS, H_LOW, X_DIM, 1);
    // Sin = h Ws^T + bs
    wmma_gemm_bias_kernel<<<dim3(MROWS / 16, Y_DIM / 64), 32, 0, stream>>>(
        hbuf, Ws, bs, Sinb, nullptr, MROWS, Y_DIM, H_LOW, 0);
    // temp_Qins -> tail of d_out
    tempq_kernel<<<B_DIM, Q_DIM, 0, stream>>>(Qrm, out + (size_t)B_DIM * L_DIM * Y_DIM);
    // fused dist/softmax/PV
    flash_kernel<<<dim3(L_DIM / 16, B_DIM), 128, (size_t)SM_FLOATS * 4, stream>>>(
        Qrm, sqb, Sinb, mask, out);
}